// RoIAlign_8873402434044
// MI455X (gfx1250) — compile-verified
//
#include <hip/hip_runtime.h>
#include <cstdint>

// RoI Align (avg, aligned=True) for MI455X / gfx1250.
// Gather-bound op: feature map (164 MB) fits in the 192 MB L2, so after the
// compulsory pass everything is L2-resident. We hide L2 gather latency with
// the CDNA5 async global->LDS gather path (ASYNCcnt), double-buffered per wave.

namespace {

constexpr int PH = 7, PW = 7, NBINS = PH * PW;
constexpr int CC = 256, HH = 200, WWD = 200;
constexpr float SCALE = 0.25f;

// Issue the 16 async gathers (4 samples x 4 bilinear corners) for one bin.
// Weights (uniform across the wave) are written into wv[0..15], folded with
// the validity mask and the 1/(sr*sr) mean factor.
__device__ __forceinline__ void issue_bin(
    int ph, int pw, float x1, float y1, float bin_w, float bin_h,
    uint64_t chan_addr, uint32_t lds_buf_addr, float* wv)
{
  // Ensure any older ds_load reads of the buffer we are about to overwrite
  // have drained before new async writes can land.
  asm volatile("s_wait_dscnt 0x0" ::: "memory");
#pragma unroll
  for (int s = 0; s < 4; ++s) {
    const int   iy = s >> 1, ix = s & 1;
    const float fy = iy ? 0.75f : 0.25f;          // (i + 0.5)/sr, sr = 2
    const float fx = ix ? 0.75f : 0.25f;
    const float y  = y1 + ((float)ph + fy) * bin_h;
    const float x  = x1 + ((float)pw + fx) * bin_w;
    const bool valid = (y >= -1.0f) && (y <= (float)HH) &&
                       (x >= -1.0f) && (x <= (float)WWD);
    const float yc = fminf(fmaxf(y, 0.0f), (float)(HH - 1));
    const float xc = fminf(fmaxf(x, 0.0f), (float)(WWD - 1));
    const int y0  = (int)floorf(yc);
    const int x0  = (int)floorf(xc);
    const int y1i = (y0 + 1 < HH - 1) ? (y0 + 1) : (HH - 1);
    const int x1i = (x0 + 1 < WWD - 1) ? (x0 + 1) : (WWD - 1);
    const float ly = yc - (float)y0, lx = xc - (float)x0;
    const float hy = 1.0f - ly,     hx = 1.0f - lx;
    const float vm = valid ? 0.25f : 0.0f;        // includes mean over 4 samples
    const float w4[4] = { hy * hx * vm, hy * lx * vm, ly * hx * vm, ly * lx * vm };
    const int   o4[4] = { y0  * WWD + x0, y0  * WWD + x1i,
                          y1i * WWD + x0, y1i * WWD + x1i };
#pragma unroll
    for (int cr = 0; cr < 4; ++cr) {
      const int k = s * 4 + cr;
      wv[k] = w4[cr];
      const uint64_t ga = chan_addr + (uint64_t)(uint32_t)(o4[cr] * 4);
      const uint32_t la = lds_buf_addr + (uint32_t)(k * 32 * 4);
      // Per-lane gather: 32 channel addresses -> 32 LDS slots, ASYNCcnt++.
      asm volatile("global_load_async_to_lds_b32 %0, %1, off"
                   :: "v"(la), "v"(ga) : "memory");
    }
  }
}

} // namespace

__global__ __launch_bounds__(256) void roi_align_async_kernel(
    const float* __restrict__ feat, const float* __restrict__ rois,
    float* __restrict__ out)
{
  // Per-wave double-buffered staging: 8 waves x 2 bufs x 16 values x 32 lanes.
  __shared__ float stage[8][2][16][32];

  const int roi  = blockIdx.x;
  const int wave = threadIdx.x >> 5;
  const int lane = threadIdx.x & 31;
  const int c    = threadIdx.x;            // channel == thread id (C = 256)

  const float* r = rois + (size_t)roi * 5;
  const int   bimg = (int)r[0];
  const float x1 = r[1] * SCALE - 0.5f;    // aligned=True offset
  const float y1 = r[2] * SCALE - 0.5f;
  const float bin_w = (r[3] * SCALE - 0.5f - x1) * (1.0f / (float)PW);
  const float bin_h = (r[4] * SCALE - 0.5f - y1) * (1.0f / (float)PH);

  const uint64_t chan_addr =
      (uint64_t)(uintptr_t)(feat + ((size_t)bimg * CC + c) * (size_t)(HH * WWD));
  // Low 32 bits of a flat LDS pointer = LDS byte offset (ISA aperture rule).
  const uint32_t lds0 = (uint32_t)(uintptr_t)(&stage[wave][0][0][lane]);
  const uint32_t lds1 = (uint32_t)(uintptr_t)(&stage[wave][1][0][lane]);

  float w0[16], w1[16];
  float* outp = out + ((size_t)roi * CC + c) * NBINS;

  // Prime the pipeline with bin 0 into buffer 0.
  issue_bin(0, 0, x1, y1, bin_w, bin_h, chan_addr, lds0, w0);

  for (int bb = 0; bb < NBINS; bb += 2) {
    // ---- even bin bb: consume buffer 0 / w0 ----
    if (bb + 1 < NBINS) {
      issue_bin((bb + 1) / PW, (bb + 1) % PW, x1, y1, bin_w, bin_h,
                chan_addr, lds1, w1);
      // 16 newer asyncs outstanding; <=16 left  =>  bin bb's 16 have landed.
      asm volatile("s_wait_asynccnt 0x10" ::: "memory");
    } else {
      asm volatile("s_wait_asynccnt 0x0" ::: "memory");
    }
    {
      float acc = 0.0f;
#pragma unroll
      for (int k = 0; k < 16; ++k)
        acc += w0[k] * stage[wave][0][k][lane];
      outp[bb] = acc;
    }

    // ---- odd bin bb+1: consume buffer 1 / w1 ----
    if (bb + 1 < NBINS) {
      if (bb + 2 < NBINS) {
        issue_bin((bb + 2) / PW, (bb + 2) % PW, x1, y1, bin_w, bin_h,
                  chan_addr, lds0, w0);
        asm volatile("s_wait_asynccnt 0x10" ::: "memory");
      } else {
        asm volatile("s_wait_asynccnt 0x0" ::: "memory");
      }
      float acc = 0.0f;
#pragma unroll
      for (int k = 0; k < 16; ++k)
        acc += w1[k] * stage[wave][1][k][lane];
      outp[bb + 1] = acc;
    }
  }
}

extern "C" void kernel_launch(void* const* d_in, const int* in_sizes, int n_in,
                              void* d_out, int out_size, void* d_ws, size_t ws_size,
                              hipStream_t stream) {
  (void)n_in; (void)out_size; (void)d_ws; (void)ws_size;
  const float* feat = (const float*)d_in[0];   // (4, 256, 200, 200) f32
  const float* rois = (const float*)d_in[1];   // (1024, 5) f32
  float*       out  = (float*)d_out;           // (1024, 256, 7, 7) f32

  const int n_rois = in_sizes[1] / 5;
  roi_align_async_kernel<<<n_rois, 256, 0, stream>>>(feat, rois, out);
}